// PAM_Module_5111011082982
// MI455X (gfx1250) — compile-verified
//
#include <hip/hip_runtime.h>

#define B_   2
#define C_   64
#define CQ_  16
#define HW_  9216
#define D_   96

#define XS_  108   // LDS stride for X tiles: 16B-aligned rows, conflict-free A-frag reads
#define QS_  100   // LDS stride for Q/K/V/attn tiles: conflict-free A-frag reads
#define WS_  68    // LDS stride for weight tiles: conflict-free A-frag reads

#define AS1 __attribute__((address_space(1)))
#define AS3 __attribute__((address_space(3)))

#ifndef __has_builtin
#define __has_builtin(x) 0
#endif

#if defined(__HIP_DEVICE_COMPILE__) &&                                   \
    __has_builtin(__builtin_amdgcn_global_load_async_to_lds_b128) &&     \
    __has_builtin(__builtin_amdgcn_s_wait_asynccnt)
#define HAS_ASYNC 1
#else
#define HAS_ASYNC 0
#endif

#if HAS_ASYNC
#define WAIT_ASYNC(n) __builtin_amdgcn_s_wait_asynccnt(n)
#else
#define WAIT_ASYNC(n) do {} while (0)
#endif

typedef float v2f __attribute__((ext_vector_type(2)));
typedef float v4f __attribute__((ext_vector_type(4)));
typedef float v8f __attribute__((ext_vector_type(8)));
typedef int   v4i __attribute__((ext_vector_type(4)));

// A-operand fragment (16x4, element [m][k] at p[m*s+k]):
//   VGPR0: lanes0-15 K=0, lanes16-31 K=2 ; VGPR1: K=1 / K=3 ; M = lane&15
static __device__ __forceinline__ v2f ld_fragA(const float* p, int s, int lane) {
  const int m = lane & 15;
  const int kh = (lane >> 4) << 1;
  v2f r;
  r.x = p[m * s + kh];
  r.y = p[m * s + kh + 1];
  return r;
}
// B-operand fragment (4x16, element [k][n] at p[k*s+n]):
//   VGPR0: lanes0-15 K=0, lanes16-31 K=2 ; VGPR1: K=1 / K=3 ; N = lane&15
// (also feeds A when the source matrix is stored transposed)
static __device__ __forceinline__ v2f ld_fragB(const float* p, int s, int lane) {
  const int n = lane & 15;
  const int kh = (lane >> 4) << 1;
  v2f r;
  r.x = p[kh * s + n];
  r.y = p[(kh + 1) * s + n];
  return r;
}
static __device__ __forceinline__ v8f wmma4(v2f a, v2f b, v8f c) {
  return __builtin_amdgcn_wmma_f32_16x16x4_f32(false, a, false, b, (short)0, c,
                                               false, false);
}

// Issue one X slab (64 x 96 fp32, row stride XS_ in LDS) as async b128 copies.
// xg points at x[b, 0, hw, 0]; per-channel stride is HW_*D_.
template <int NT>
static __device__ __forceinline__ void issue_x_tile(const float* __restrict__ xg,
                                                    float* __restrict__ dst,
                                                    int tid) {
  for (int j = tid; j < C_ * 24; j += NT) {
    const int c = j / 24, d4 = (j % 24) * 4;
#if HAS_ASYNC
    __builtin_amdgcn_global_load_async_to_lds_b128(
        (AS1 v4i*)(AS1 void*)(xg + (size_t)c * HW_ * D_ + d4),
        (AS3 v4i*)(AS3 void*)(dst + c * XS_ + d4), 0, 0);
#else
    *(v4f*)&dst[c * XS_ + d4] = *(const v4f*)&xg[(size_t)c * HW_ * D_ + d4];
#endif
  }
}

// ---------------------------------------------------------------------------
// Kernel 1: energy[b,d,e] += sum over (o,hw) of q[(o,hw),d] * k[(o,hw),e]
// fused with the q/k projections. 6 waves; wave w owns d-stripe [16w,16w+16).
// Double-buffered async X tiles.
// ---------------------------------------------------------------------------
__global__ __launch_bounds__(192) void pam_energy_kernel(
    const float* __restrict__ x, const float* __restrict__ Wq,
    const float* __restrict__ bq, const float* __restrict__ Wk,
    const float* __restrict__ bk, float* __restrict__ energy) {
  __shared__ float Xs[2][C_ * XS_];
  __shared__ float Qs[CQ_ * QS_];
  __shared__ float Ks[CQ_ * QS_];
  __shared__ float Wqs[CQ_ * WS_];
  __shared__ float Wks[CQ_ * WS_];
  __shared__ float bqs[CQ_];
  __shared__ float bks[CQ_];

  const int b = blockIdx.x / 192;
  const int chunk = blockIdx.x % 192;
  const int tid = threadIdx.x;
  const int wave = tid >> 5;
  const int lane = tid & 31;

  for (int i = tid; i < CQ_ * C_; i += 192) {
    Wqs[(i >> 6) * WS_ + (i & 63)] = Wq[i];
    Wks[(i >> 6) * WS_ + (i & 63)] = Wk[i];
  }
  if (tid < CQ_) { bqs[tid] = bq[tid]; bks[tid] = bk[tid]; }

  const float* xb = x + (size_t)b * C_ * HW_ * D_;
  const int hw0 = chunk * 48;
  issue_x_tile<192>(xb + (size_t)hw0 * D_, Xs[0], tid);  // prologue: tile 0

  const v8f vzero = {};
  v8f E[6];
#pragma unroll
  for (int t = 0; t < 6; ++t) E[t] = vzero;

  const int mBase = (lane >> 4) * 8;  // C/D layout: VGPR r holds M=r (+8 hi half)
  const int nCol = lane & 15;

  for (int it = 0; it < 48; ++it) {
    const float* Xcur = Xs[it & 1];
    if (it + 1 < 48) {
      issue_x_tile<192>(xb + (size_t)(hw0 + it + 1) * D_, Xs[(it + 1) & 1], tid);
      WAIT_ASYNC(8);  // 8 copies/thread per tile: previous tile fully landed
    } else {
      WAIT_ASYNC(0);
    }
    __syncthreads();

    // Q16 = Wq @ X + bq, K16 = Wk @ X + bk (wave w: d-cols [16w,16w+16))
    v8f q = vzero, k = vzero;
#pragma unroll
    for (int kk = 0; kk < 16; ++kk) {
      const v2f aq = ld_fragA(&Wqs[kk * 4], WS_, lane);
      const v2f ak = ld_fragA(&Wks[kk * 4], WS_, lane);
      const v2f bx = ld_fragB(&Xcur[kk * 4 * XS_ + wave * 16], XS_, lane);
      q = wmma4(aq, bx, q);
      k = wmma4(ak, bx, k);
    }
#pragma unroll
    for (int r = 0; r < 8; ++r) {
      Qs[(mBase + r) * QS_ + wave * 16 + nCol] = q[r] + bqs[mBase + r];
      Ks[(mBase + r) * QS_ + wave * 16 + nCol] = k[r] + bks[mBase + r];
    }
    __syncthreads();

    // E += Q16^T @ K16  (A-fragment hoisted over the 6 e-tiles)
#pragma unroll
    for (int kk = 0; kk < 4; ++kk) {
      const v2f a = ld_fragB(&Qs[kk * 4 * QS_ + wave * 16], QS_, lane);
#pragma unroll
      for (int et = 0; et < 6; ++et) {
        const v2f bb = ld_fragB(&Ks[kk * 4 * QS_ + et * 16], QS_, lane);
        E[et] = wmma4(a, bb, E[et]);
      }
    }
  }

  float* eg = energy + (size_t)b * D_ * D_;
#pragma unroll
  for (int et = 0; et < 6; ++et) {
#pragma unroll
    for (int r = 0; r < 8; ++r) {
      const int d = wave * 16 + mBase + r;
      const int e = et * 16 + nCol;
      atomicAdd(&eg[d * D_ + e], E[et][r]);
    }
  }
}

// ---------------------------------------------------------------------------
// Kernel 2: row softmax over e (B*96 rows of 96), one wave per row.
// ---------------------------------------------------------------------------
__global__ __launch_bounds__(32) void pam_softmax_kernel(
    const float* __restrict__ energy, float* __restrict__ attn) {
  const int row = blockIdx.x;
  const int lane = threadIdx.x;
  const float* er = energy + (size_t)row * D_;
  const float v0 = er[lane], v1 = er[lane + 32], v2 = er[lane + 64];
  float m = fmaxf(v0, fmaxf(v1, v2));
#pragma unroll
  for (int off = 16; off > 0; off >>= 1) m = fmaxf(m, __shfl_xor(m, off, 32));
  const float e0 = __expf(v0 - m), e1 = __expf(v1 - m), e2 = __expf(v2 - m);
  float s = e0 + e1 + e2;
#pragma unroll
  for (int off = 16; off > 0; off >>= 1) s += __shfl_xor(s, off, 32);
  const float inv = 1.0f / s;
  float* ar = attn + (size_t)row * D_;
  ar[lane] = e0 * inv;
  ar[lane + 32] = e1 * inv;
  ar[lane + 64] = e2 * inv;
}

// ---------------------------------------------------------------------------
// Kernel 3: per hw slab: V = Wv@X + bv ; O = V @ attn^T ; out = gamma*O + X
// 8 waves; wave w owns c-stripe mt=w/2 and d-cols [(w&1)*48, +48).
// Double-buffered async X tiles; staged b128 non-temporal output stores.
// ---------------------------------------------------------------------------
__global__ __launch_bounds__(256) void pam_out_kernel(
    const float* __restrict__ x, const float* __restrict__ Wv,
    const float* __restrict__ bv, const float* __restrict__ attn,
    const float* __restrict__ gamma, float* __restrict__ out) {
  __shared__ float Xs[2][C_ * XS_];
  __shared__ float Vs[C_ * QS_];
  __shared__ float As[D_ * QS_];  // attn[d][e] row-major
  __shared__ float Wvs[C_ * WS_];
  __shared__ float bvs[C_];

  const int b = blockIdx.x / 288;
  const int chunk = blockIdx.x % 288;
  const int tid = threadIdx.x;
  const int wave = tid >> 5;
  const int lane = tid & 31;

  for (int i = tid; i < C_ * C_; i += 256)
    Wvs[(i >> 6) * WS_ + (i & 63)] = Wv[i];
  if (tid < C_) bvs[tid] = bv[tid];
  const float* ag = attn + (size_t)b * D_ * D_;
  for (int i = tid; i < D_ * D_; i += 256)
    As[(i / D_) * QS_ + (i % D_)] = ag[i];
  const float g = gamma[0];

  const float* xb = x + (size_t)b * C_ * HW_ * D_;
  float* ob = out + (size_t)b * C_ * HW_ * D_;
  const int hw0 = chunk * 32;
  issue_x_tile<256>(xb + (size_t)hw0 * D_, Xs[0], tid);  // prologue: tile 0

  const int mBase = (lane >> 4) * 8;
  const int nCol = lane & 15;
  const int mt = wave >> 1;        // c-tile (0..3)
  const int ntB = (wave & 1) * 3;  // first d-tile of this wave's 3
  const v8f vzero = {};

  for (int it = 0; it < 32; ++it) {
    const int hw = hw0 + it;
    float* Xcur = Xs[it & 1];
    if (it + 1 < 32) {
      issue_x_tile<256>(xb + (size_t)(hw + 1) * D_, Xs[(it + 1) & 1], tid);
      WAIT_ASYNC(6);  // 6 copies/thread per tile
    } else {
      WAIT_ASYNC(0);
    }
    __syncthreads();

    // V = Wv @ X + bv  (A-fragment shared across the wave's 3 d-tiles)
    {
      v8f a0 = vzero, a1 = vzero, a2 = vzero;
#pragma unroll
      for (int kk = 0; kk < 16; ++kk) {
        const v2f a = ld_fragA(&Wvs[mt * 16 * WS_ + kk * 4], WS_, lane);
        const v2f b0 = ld_fragB(&Xcur[kk * 4 * XS_ + (ntB + 0) * 16], XS_, lane);
        a0 = wmma4(a, b0, a0);
        const v2f b1 = ld_fragB(&Xcur[kk * 4 * XS_ + (ntB + 1) * 16], XS_, lane);
        a1 = wmma4(a, b1, a1);
        const v2f b2 = ld_fragB(&Xcur[kk * 4 * XS_ + (ntB + 2) * 16], XS_, lane);
        a2 = wmma4(a, b2, a2);
      }
#pragma unroll
      for (int r = 0; r < 8; ++r) {
        const int c = mt * 16 + mBase + r;
        const float bb = bvs[c];
        Vs[c * QS_ + (ntB + 0) * 16 + nCol] = a0[r] + bb;
        Vs[c * QS_ + (ntB + 1) * 16 + nCol] = a1[r] + bb;
        Vs[c * QS_ + (ntB + 2) * 16 + nCol] = a2[r] + bb;
      }
    }
    __syncthreads();

    // O[c,d] = sum_e V[c,e]*attn[d,e]; write gamma*O + X in place into Xcur
    {
      v8f a0 = vzero, a1 = vzero, a2 = vzero;
#pragma unroll
      for (int kk = 0; kk < 24; ++kk) {
        const v2f a = ld_fragA(&Vs[mt * 16 * QS_ + kk * 4], QS_, lane);
        // B[k=e][n=d] = attn[d][e]: row-major attn via the A-pattern loader
        const v2f b0 = ld_fragA(&As[(ntB + 0) * 16 * QS_ + kk * 4], QS_, lane);
        a0 = wmma4(a, b0, a0);
        const v2f b1 = ld_fragA(&As[(ntB + 1) * 16 * QS_ + kk * 4], QS_, lane);
        a1 = wmma4(a, b1, a1);
        const v2f b2 = ld_fragA(&As[(ntB + 2) * 16 * QS_ + kk * 4], QS_, lane);
        a2 = wmma4(a, b2, a2);
      }
#pragma unroll
      for (int r = 0; r < 8; ++r) {
        float* row = &Xcur[(mt * 16 + mBase + r) * XS_];
        const int d0 = (ntB + 0) * 16 + nCol;
        const int d1 = (ntB + 1) * 16 + nCol;
        const int d2 = (ntB + 2) * 16 + nCol;
        row[d0] = g * a0[r] + row[d0];
        row[d1] = g * a1[r] + row[d1];
        row[d2] = g * a2[r] + row[d2];
      }
    }
    __syncthreads();

    // Coalesced b128 non-temporal stores of the finished slab
    for (int j = tid; j < C_ * 24; j += 256) {
      const int c = j / 24, d4 = (j % 24) * 4;
      const v4f v = *(const v4f*)&Xcur[c * XS_ + d4];
      __builtin_nontemporal_store(v, (v4f*)&ob[((size_t)c * HW_ + hw) * D_ + d4]);
    }
    __syncthreads();  // Xcur reads done before next iter's prefetch re-fills it
  }
}

extern "C" void kernel_launch(void* const* d_in, const int* in_sizes, int n_in,
                              void* d_out, int out_size, void* d_ws,
                              size_t ws_size, hipStream_t stream) {
  const float* x = (const float*)d_in[0];
  const float* Wq = (const float*)d_in[1];
  const float* bq = (const float*)d_in[2];
  const float* Wk = (const float*)d_in[3];
  const float* bk = (const float*)d_in[4];
  const float* Wv = (const float*)d_in[5];
  const float* bv = (const float*)d_in[6];
  const float* gamma = (const float*)d_in[7];
  float* out = (float*)d_out;

  float* energy = (float*)d_ws;                 // B*96*96
  float* attn = energy + (size_t)B_ * D_ * D_;  // B*96*96

  (void)hipMemsetAsync(energy, 0, (size_t)B_ * D_ * D_ * sizeof(float), stream);
  pam_energy_kernel<<<B_ * 192, 192, 0, stream>>>(x, Wq, bq, Wk, bk, energy);
  pam_softmax_kernel<<<B_ * D_, 32, 0, stream>>>(energy, attn);
  pam_out_kernel<<<B_ * 288, 256, 0, stream>>>(x, Wv, bv, attn, gamma, out);

  (void)in_sizes; (void)n_in; (void)out_size; (void)ws_size;
}